// MixHopConv_27994596835371
// MI455X (gfx1250) — compile-verified
//
#include <hip/hip_runtime.h>

typedef float v2f __attribute__((ext_vector_type(2)));
typedef float v8f __attribute__((ext_vector_type(8)));

#define F 256  // F_IN == F_OUT == 256

// ---------------------------------------------------------------------------
// Zero-fill (float4 grid-stride). Needed because SpMM accumulates with atomics
// into poisoned d_out / d_ws regions.
// ---------------------------------------------------------------------------
__global__ void mixhop_zero_f4(float4* __restrict__ p, long long n4) {
  long long i = (long long)blockIdx.x * blockDim.x + threadIdx.x;
  long long stride = (long long)gridDim.x * blockDim.x;
  float4 z = {0.f, 0.f, 0.f, 0.f};
  for (; i < n4; i += stride) p[i] = z;
}

// ---------------------------------------------------------------------------
// GEMM: out[n, coff + c] = sum_k x[n,k] * W[k,c] + bias[c]
// Block = 256 threads = 8 waves. Each wave computes a 16x64 output strip
// (4 accumulators of 16x16) via V_WMMA_F32_16X16X4_F32, so each A fragment
// feeds 4 WMMAs (4x better x-reuse than 1 tile/wave).
// The 256x64 f32 W tile (64 KB) is staged into LDS with per-lane
// GLOBAL_LOAD_ASYNC_TO_LDS_B128 (ASYNCcnt path), then all waves sync.
//
// A-frag (16x4 f32, 2 VGPRs/lane): lanes 0-15: M=lane, K=k..k+1;
// lanes 16-31: M=lane-16, K=k+2..k+3  -> per-lane contiguous float2 load.
// B-frag (4x16 f32): VGPR0 = row k (lanes 0-15) / row k+2 (lanes 16-31),
// VGPR1 = row k+1 / k+3.
// C/D: VGPR v -> M = v (lanes 0-15) or v+8 (lanes 16-31), N = lane&15.
// ---------------------------------------------------------------------------
__global__ void __launch_bounds__(256) mixhop_gemm_wmma_f32(
    const float* __restrict__ x, const float* __restrict__ W,
    const float* __restrict__ bias, float* __restrict__ out,
    int N, int ldo, int coff) {
  __shared__ float Bs[F * 64];  // [k][c] : 256 rows x 64 cols = 64 KB

  const int tid     = threadIdx.x;
  const int colBase = blockIdx.x * 64;

  // ---- Stage W column tile [256 x 64] into LDS via async-to-LDS b128 ----
  // Each thread moves 16 float4 (256 B); 256 threads cover the 64 KB tile.
  {
    const float* wtile = W + colBase;  // row k: wtile[k*F + c], c in [0,64)
    unsigned ldsBase =
        (unsigned)(size_t)(__attribute__((address_space(3))) void*)Bs;
#pragma unroll
    for (int t = 0; t < 16; ++t) {
      const int j  = t * 256 + tid;  // float4 index within the tile
      const int k  = j >> 4;         // 16 float4 per 64-col row
      const int c4 = j & 15;
      const unsigned goff  = (unsigned)(((k * F) + (c4 << 2)) << 2);  // bytes
      const unsigned laddr = ldsBase + ((unsigned)j << 4);
      asm volatile("global_load_async_to_lds_b128 %0, %1, %2"
                   :: "v"(laddr), "v"(goff), "s"(wtile)
                   : "memory");
    }
    asm volatile("s_wait_asynccnt 0x0" ::: "memory");
  }
  __syncthreads();

  const int wave = tid >> 5;
  const int lane = tid & 31;
  const int lo   = lane & 15;
  const int hi   = lane >> 4;  // 0: K pair {0,1}; 1: K pair {2,3}

  const int row0 = (blockIdx.y * 8 + wave) * 16;
  if (row0 >= N) return;  // wave-uniform -> EXEC all-ones inside active waves

  const float* aptr = x + (size_t)(row0 + lo) * F + (hi << 1);

  v8f acc[4];
#pragma unroll
  for (int t = 0; t < 4; ++t) {
    const float bv = bias[colBase + t * 16 + lo];  // bias depends on col only
    acc[t] = (v8f){bv, bv, bv, bv, bv, bv, bv, bv};
  }

#pragma unroll 4
  for (int k = 0; k < F; k += 4) {
    const v2f a = *(const v2f*)(aptr + k);
#pragma unroll
    for (int t = 0; t < 4; ++t) {
      v2f b;
      b.x = Bs[(k + 0 + (hi << 1)) * 64 + t * 16 + lo];
      b.y = Bs[(k + 1 + (hi << 1)) * 64 + t * 16 + lo];
      // 8 args: (neg_a, A, neg_b, B, c_mod, C, reuse_a, reuse_b)
      acc[t] = __builtin_amdgcn_wmma_f32_16x16x4_f32(
          false, a, false, b, (short)0, acc[t], false, false);
    }
  }

  float* obase = out + (size_t)row0 * ldo + coff + colBase;
#pragma unroll
  for (int t = 0; t < 4; ++t) {
#pragma unroll
    for (int v = 0; v < 8; ++v) {
      obase[(size_t)(v + (hi << 3)) * ldo + t * 16 + lo] = acc[t][v];
    }
  }
}

// ---------------------------------------------------------------------------
// SpMM scatter: dst[row[e], coff + :] += vals[e] * src[col[e], :]
// One wave per edge (grid-stride). Lane reads 8 f32 via two float4 loads
// (coalesced 512B per half-row), then 8 native f32 atomics into L2.
// Prefetch next edge's source row (global_prefetch_b8).
// ---------------------------------------------------------------------------
__global__ void __launch_bounds__(256) mixhop_spmm_atomic(
    const long long* __restrict__ row, const long long* __restrict__ col,
    const float* __restrict__ vals, const float* __restrict__ src,
    float* __restrict__ dst, int E, int ldd, int coff) {
  const int lane  = threadIdx.x & 31;
  const int wave  = blockIdx.x * (blockDim.x >> 5) + (threadIdx.x >> 5);
  const int nwave = gridDim.x * (blockDim.x >> 5);

  for (int e = wave; e < E; e += nwave) {
    const long long r = row[e];
    const long long c = col[e];
    const float     v = vals[e];

    // Prefetch next iteration's gather row while this one is in flight.
    const int en = e + nwave;
    if (en < E) {
      const long long cn = col[en];
      __builtin_prefetch(src + (size_t)cn * F + lane * 8, 0, 0);
    }

    const float4* s  = (const float4*)(src + (size_t)c * F);
    const float4  f0 = s[lane];        // cols lane*4 .. lane*4+3
    const float4  f1 = s[lane + 32];   // cols 128 + lane*4 ..

    float* d0 = dst + (size_t)r * ldd + coff + lane * 4;
    unsafeAtomicAdd(d0 + 0, v * f0.x);
    unsafeAtomicAdd(d0 + 1, v * f0.y);
    unsafeAtomicAdd(d0 + 2, v * f0.z);
    unsafeAtomicAdd(d0 + 3, v * f0.w);
    float* d1 = d0 + 128;
    unsafeAtomicAdd(d1 + 0, v * f1.x);
    unsafeAtomicAdd(d1 + 1, v * f1.y);
    unsafeAtomicAdd(d1 + 2, v * f1.z);
    unsafeAtomicAdd(d1 + 3, v * f1.w);
  }
}

// ---------------------------------------------------------------------------
// Orchestration
// inputs: x[N,256] f32, W[3,256,256] f32, b[3,256] f32,
//         row[E] i64, col[E] i64, vals[E] f32
// output: [N, 768] f32 = concat(h0, A@h1, A@A@h2) with h_j = x@W[j]+b[j]
// ---------------------------------------------------------------------------
extern "C" void kernel_launch(void* const* d_in, const int* in_sizes, int n_in,
                              void* d_out, int out_size, void* d_ws, size_t ws_size,
                              hipStream_t stream) {
  const float*     x    = (const float*)d_in[0];
  const float*     W    = (const float*)d_in[1];
  const float*     bias = (const float*)d_in[2];
  const long long* row  = (const long long*)d_in[3];
  const long long* col  = (const long long*)d_in[4];
  const float*     vals = (const float*)d_in[5];
  float*           out  = (float*)d_out;

  const int N = in_sizes[0] / F;
  const int E = in_sizes[3];
  const int LDO = 3 * F;  // 768

  float* tmpA = (float*)d_ws;                 // N*256 f32
  float* tmpB = tmpA + (size_t)N * F;         // N*256 f32

  // Zero atomic-accumulation targets.
  mixhop_zero_f4<<<2048, 256, 0, stream>>>((float4*)out,
                                           ((long long)N * LDO) / 4);
  mixhop_zero_f4<<<2048, 256, 0, stream>>>((float4*)tmpB,
                                           ((long long)N * F) / 4);

  const dim3 gemmGrid(F / 64, ((N + 15) / 16 + 7) / 8);
  const int spmmBlocks = 4096;  // 32768 waves, grid-stride over edges

  // hop 0: out[:, 0:256] = x @ W0 + b0
  mixhop_gemm_wmma_f32<<<gemmGrid, 256, 0, stream>>>(
      x, W + 0 * F * F, bias + 0 * F, out, N, LDO, 0);

  // hop 1: tmpA = x @ W1 + b1 ; out[:, 256:512] += A @ tmpA
  mixhop_gemm_wmma_f32<<<gemmGrid, 256, 0, stream>>>(
      x, W + 1 * F * F, bias + 1 * F, tmpA, N, F, 0);
  mixhop_spmm_atomic<<<spmmBlocks, 256, 0, stream>>>(
      row, col, vals, tmpA, out, E, LDO, F);

  // hop 2: tmpA = x @ W2 + b2 ; tmpB = A @ tmpA ; out[:, 512:768] += A @ tmpB
  mixhop_gemm_wmma_f32<<<gemmGrid, 256, 0, stream>>>(
      x, W + 2 * F * F, bias + 2 * F, tmpA, N, F, 0);
  mixhop_spmm_atomic<<<spmmBlocks, 256, 0, stream>>>(
      row, col, vals, tmpA, tmpB, E, F, 0);
  mixhop_spmm_atomic<<<spmmBlocks, 256, 0, stream>>>(
      row, col, vals, tmpB, out, E, LDO, 2 * F);
}